// BatchAllTtripletLoss_72464688218879
// MI455X (gfx1250) — compile-verified
//
#include <hip/hip_runtime.h>

#define BN 512
#define DN 256
#define MARGIN_F 1.0f
#define EPS_F 1e-8f

typedef __attribute__((ext_vector_type(2))) float v2f;
typedef __attribute__((ext_vector_type(8))) float v8f;

// ---------------------------------------------------------------------------
// Kernel 0: row squared norms  sq[i] = sum_k X[i,k]^2
// ---------------------------------------------------------------------------
__global__ __launch_bounds__(256) void sqnorm_kernel(const float* __restrict__ x,
                                                     float* __restrict__ sq) {
  int i = blockIdx.x * blockDim.x + threadIdx.x;
  if (i < BN) {
    const float* row = x + i * DN;
    float s = 0.0f;
#pragma unroll 4
    for (int k = 0; k < DN; ++k) s += row[k] * row[k];
    sq[i] = s;
  }
}

// ---------------------------------------------------------------------------
// Kernel 1: Euclidean distance matrix via f32 WMMA (one wave per 16x16 tile).
//
// V_WMMA_F32_16X16X4_F32:  D(16x16) = A(16x4) x B(4x16) + C
//   A layout (ISA 7.12.2, 32-bit A 16x4): lane L holds row M = L%16,
//     VGPR v holds K = (L/16)*2 + v  -> contiguous float2 per lane.
//   B is X^T tile of the symmetric Gram product; mirrored load.
//   C/D layout: lane L, VGPR r -> M = r + (L/16)*8, N = L%16.
// ---------------------------------------------------------------------------
__global__ __launch_bounds__(256) void dist_kernel(const float* __restrict__ x,
                                                   const float* __restrict__ sq,
                                                   float* __restrict__ dist) {
  const int lane = threadIdx.x & 31;
  const int wave = threadIdx.x >> 5;
  const int tile = blockIdx.x * 8 + wave;  // 0..1023 -> 32x32 tiles of 16x16
  const int tm = tile >> 5;
  const int tn = tile & 31;
  const int half = lane >> 4;  // 0 or 1
  const int l16 = lane & 15;

  const float* pa = x + (tm * 16 + l16) * DN + half * 2;
  const float* pb = x + (tn * 16 + l16) * DN + half * 2;

  v8f acc = {};
  for (int k = 0; k < DN; k += 4) {
    v2f a = *(const v2f*)(pa + k);
    v2f b = *(const v2f*)(pb + k);
    // 8 args: (neg_a, A, neg_b, B, c_mod, C, reuse_a, reuse_b)
    acc = __builtin_amdgcn_wmma_f32_16x16x4_f32(false, a, false, b,
                                                (short)0, acc, false, false);
  }

  const int j = tn * 16 + l16;
  const float sqj = sq[j];
#pragma unroll
  for (int r = 0; r < 8; ++r) {
    const int i = tm * 16 + half * 8 + r;
    float d2 = sq[i] - 2.0f * acc[r] + sqj;
    d2 = d2 > 0.0f ? d2 : 0.0f;                      // relu
    dist[i * BN + j] = (d2 == 0.0f) ? 0.0f : sqrtf(d2);  // zero-mask semantics
  }
}

// ---------------------------------------------------------------------------
// Kernel 2: batch-all triplet scan, one block per anchor.
// Mask algebra: n-validity (labels[n]!=la && groups[n]==ga) already implies
// n!=a and n!=p; only p!=a needs an explicit distinct check.
// ---------------------------------------------------------------------------
__global__ __launch_bounds__(256) void triplet_kernel(const float* __restrict__ dist,
                                                      const int* __restrict__ labels,
                                                      const int* __restrict__ groups,
                                                      float* __restrict__ bsum,
                                                      unsigned int* __restrict__ bcnt) {
  __shared__ float srow[BN];
  __shared__ unsigned char nval[BN];
  __shared__ float rsum[256];
  __shared__ unsigned int rcnt[256];

  const int a = blockIdx.x;
  const int tid = threadIdx.x;
  const int la = labels[a];
  const int ga = groups[a];

  for (int n = tid; n < BN; n += 256) {
    srow[n] = dist[a * BN + n];
    nval[n] = (unsigned char)((labels[n] != la) && (groups[n] == ga));
  }
  __syncthreads();

  float sum = 0.0f;
  unsigned int cnt = 0u;
  for (int p = tid; p < BN; p += 256) {
    const bool pv = (p != a) && (labels[p] == la) && (groups[p] == ga);
    if (!pv) continue;
    const float base = srow[p] + MARGIN_F;
    for (int n = 0; n < BN; ++n) {
      if (!nval[n]) continue;
      const float v = base - srow[n];
      if (v > 0.0f) {
        sum += v;
        if (v > EPS_F) ++cnt;
      }
    }
  }

  rsum[tid] = sum;
  rcnt[tid] = cnt;
  __syncthreads();
  for (int s = 128; s > 0; s >>= 1) {
    if (tid < s) {
      rsum[tid] += rsum[tid + s];
      rcnt[tid] += rcnt[tid + s];
    }
    __syncthreads();
  }
  if (tid == 0) {
    bsum[a] = rsum[0];
    bcnt[a] = rcnt[0];
  }
}

// ---------------------------------------------------------------------------
// Kernel 3: final deterministic reduction of 512 partials.
// ---------------------------------------------------------------------------
__global__ __launch_bounds__(256) void final_kernel(const float* __restrict__ bsum,
                                                    const unsigned int* __restrict__ bcnt,
                                                    float* __restrict__ out) {
  __shared__ float rs[256];
  __shared__ unsigned int rc[256];
  const int tid = threadIdx.x;
  rs[tid] = bsum[tid] + bsum[tid + 256];
  rc[tid] = bcnt[tid] + bcnt[tid + 256];
  __syncthreads();
  for (int s = 128; s > 0; s >>= 1) {
    if (tid < s) {
      rs[tid] += rs[tid + s];
      rc[tid] += rc[tid + s];
    }
    __syncthreads();
  }
  if (tid == 0) out[0] = rs[0] / ((float)rc[0] + EPS_F);
}

extern "C" void kernel_launch(void* const* d_in, const int* in_sizes, int n_in,
                              void* d_out, int out_size, void* d_ws, size_t ws_size,
                              hipStream_t stream) {
  (void)in_sizes; (void)n_in; (void)out_size; (void)ws_size;
  const float* x      = (const float*)d_in[0];   // (512, 256) f32
  const int* labels   = (const int*)d_in[1];     // (512,) i32
  const int* groups   = (const int*)d_in[2];     // (512,) i32
  float* out          = (float*)d_out;           // scalar f32

  char* ws = (char*)d_ws;
  float* dist        = (float*)ws;                                   // 512*512*4 = 1 MB
  float* sq          = (float*)(ws + (size_t)BN * BN * 4);           // 2 KB
  float* bsum        = (float*)(ws + (size_t)BN * BN * 4 + BN * 4);  // 2 KB
  unsigned int* bcnt = (unsigned int*)(ws + (size_t)BN * BN * 4 + 2 * BN * 4);

  sqnorm_kernel<<<2, 256, 0, stream>>>(x, sq);
  dist_kernel<<<128, 256, 0, stream>>>(x, sq, dist);
  triplet_kernel<<<BN, 256, 0, stream>>>(dist, labels, groups, bsum, bcnt);
  final_kernel<<<1, 256, 0, stream>>>(bsum, bcnt, out);
}